// MBGCN_18459769439003
// MI455X (gfx1250) — compile-verified
//
#include <hip/hip_runtime.h>

#define U_N 100000
#define I_N 50000
#define E_N 64
#define R_N 3
#define PE_N 16
#define FH_N 64
#define B_N 512
#define K_N 100
#define NNZ_UI_N 500000
#define NNZ_II_N 500000
#define NNZ_UU_N 500000
#define NNZ_TR_N 1000000
#define EPS_F 1e-8f
#define LAMB_F 0.5f
#define L2N_F 1e-4f

typedef __attribute__((ext_vector_type(2))) float v2f;
typedef __attribute__((ext_vector_type(8))) float v8f;

static __device__ __forceinline__ v8f wmma4(v2f a, v2f b, v8f c) {
  // D = A(16x4 f32) * B(4x16 f32) + C(16x16 f32)
  return __builtin_amdgcn_wmma_f32_16x16x4_f32(false, a, false, b, (short)0, c,
                                               false, false);
}

static __device__ __forceinline__ float wave_sum(float v) {
#pragma unroll
  for (int o = 16; o > 0; o >>= 1) v += __shfl_down(v, o, 32);
  return v;
}

// ---------------------------------------------------------------------------
// slot[user[b]] = b  (batch membership; duplicate uids: any winner is fine,
// all per-user values depend only on the uid)
// ---------------------------------------------------------------------------
__global__ void k_slot(const int* __restrict__ user, int* __restrict__ slot, int n) {
  int t = blockIdx.x * blockDim.x + threadIdx.x;
  if (t < n) slot[user[t]] = t;
}

// ---------------------------------------------------------------------------
// fused_item = [item_emb | cat_tab[category] | price_tab[price]] @ W^T + b
// W: (64,96) row-major; B-matrix[k][n] = W[n*96+k]. nrows multiple of 16.
// ---------------------------------------------------------------------------
__global__ void k_fused_item(const float* __restrict__ item_emb,
                             const float* __restrict__ cat_tab,
                             const float* __restrict__ price_tab,
                             const int* __restrict__ category,
                             const int* __restrict__ price,
                             const float* __restrict__ W,
                             const float* __restrict__ bias,
                             float* __restrict__ out, int nrows) {
  const int wave = (blockIdx.x * blockDim.x + threadIdx.x) >> 5;
  const int lane = threadIdx.x & 31;
  const int m0 = (wave >> 2) << 4;
  if (m0 >= nrows) return;
  const int n0 = (wave & 3) << 4;
  const int half = lane >> 4;
  const int l = lane & 15;
  const int am = m0 + l;
  const float* erow = item_emb + (size_t)am * 64;
  const float* crow = cat_tab + (size_t)category[am] * 16;
  const float* prow = price_tab + (size_t)price[am] * 16;
  const float* wrow = W + (size_t)(n0 + l) * 96;
  v8f c = {};
#pragma unroll
  for (int k0 = 0; k0 < 96; k0 += 4) {
    const int ka = k0 + half * 2;
    v2f a, b;
    if (k0 < 64)      { a.x = erow[ka];      a.y = erow[ka + 1]; }
    else if (k0 < 80) { a.x = crow[ka - 64]; a.y = crow[ka - 63]; }
    else              { a.x = prow[ka - 80]; a.y = prow[ka - 79]; }
    b.x = wrow[ka];
    b.y = wrow[ka + 1];
    c = wmma4(a, b, c);
  }
  const float bv = bias[n0 + l];
  float* orow = out + (size_t)(m0 + half * 8) * 64 + n0 + l;
#pragma unroll
  for (int v = 0; v < 8; ++v) orow[(size_t)v * 64] = c[v] + bv;
}

// ---------------------------------------------------------------------------
// fused_user = [user_emb | handcrafted] @ W^T + b ; W: (64,128)
// ---------------------------------------------------------------------------
__global__ void k_fused_user(const float* __restrict__ user_emb,
                             const float* __restrict__ hand,
                             const float* __restrict__ W,
                             const float* __restrict__ bias,
                             float* __restrict__ out, int nrows) {
  const int wave = (blockIdx.x * blockDim.x + threadIdx.x) >> 5;
  const int lane = threadIdx.x & 31;
  const int m0 = (wave >> 2) << 4;
  if (m0 >= nrows) return;
  const int n0 = (wave & 3) << 4;
  const int half = lane >> 4;
  const int l = lane & 15;
  const int am = m0 + l;
  const float* erow = user_emb + (size_t)am * 64;
  const float* hrow = hand + (size_t)am * 64;
  const float* wrow = W + (size_t)(n0 + l) * 128;
  v8f c = {};
#pragma unroll
  for (int k0 = 0; k0 < 128; k0 += 4) {
    const int ka = k0 + half * 2;
    v2f a, b;
    if (k0 < 64) { a.x = erow[ka];      a.y = erow[ka + 1]; }
    else         { a.x = hrow[ka - 64]; a.y = hrow[ka - 63]; }
    b.x = wrow[ka];
    b.y = wrow[ka + 1];
    c = wmma4(a, b, c);
  }
  const float bv = bias[n0 + l];
  float* orow = out + (size_t)(m0 + half * 8) * 64 + n0 + l;
#pragma unroll
  for (int v = 0; v < 8; ++v) orow[(size_t)v * 64] = c[v] + bv;
}

// ---------------------------------------------------------------------------
// Unfiltered segment-sum SpMM, grid-stride wave-per-edge, 64 cols, with
// global_prefetch look-ahead on the next gather row.
// acc[rows[e]*64 + j] += vals[e] * X[cols[e]*64 + j]
// ---------------------------------------------------------------------------
__global__ void k_spmm(const int* __restrict__ rows, const int* __restrict__ cols,
                       const float* __restrict__ vals, const float* __restrict__ X,
                       float* __restrict__ acc, int nnz) {
  const int lane = threadIdx.x & 31;
  const int wave = (blockIdx.x * blockDim.x + threadIdx.x) >> 5;
  const int nw = (gridDim.x * blockDim.x) >> 5;
  for (int e = wave; e < nnz; e += nw) {
    const int r = rows[e];
    const int cidx = cols[e];
    const float v = vals[e];
    const int en = e + nw;
    if (en < nnz && (lane & 15) == 0) {
      // prefetch both 128B lines of the next 256B gather row
      __builtin_prefetch(&X[(size_t)cols[en] * 64 + (lane & 16) * 2], 0, 3);
    }
    const float* src = X + (size_t)cidx * 64;
    float* dst = acc + (size_t)r * 64;
    atomicAdd(&dst[lane], v * src[lane]);
    atomicAdd(&dst[lane + 32], v * src[lane + 32]);
  }
}

// ---------------------------------------------------------------------------
// Edge compaction: thread-per-edge; keep edges whose row is a batch user.
// ---------------------------------------------------------------------------
__global__ void k_filter(const int* __restrict__ rows, const int* __restrict__ slot,
                         int* __restrict__ cnt, int* __restrict__ elist, int nnz) {
  const int t = blockIdx.x * blockDim.x + threadIdx.x;
  if (t >= nnz) return;
  if (slot[rows[t]] >= 0) {
    const int idx = atomicAdd(cnt, 1);
    elist[idx] = t;
  }
}

// ---------------------------------------------------------------------------
// List SpMM, 64 cols (uu -> UPR): wave per compacted edge, grid-stride.
// ---------------------------------------------------------------------------
__global__ void k_spmm_list64(const int* __restrict__ elist, const int* __restrict__ cnt,
                              const int* __restrict__ rows, const int* __restrict__ cols,
                              const float* __restrict__ vals, const int* __restrict__ slot,
                              const float* __restrict__ X, float* __restrict__ acc) {
  const int lane = threadIdx.x & 31;
  const int wave = (blockIdx.x * blockDim.x + threadIdx.x) >> 5;
  const int nw = (gridDim.x * blockDim.x) >> 5;
  const int n = *cnt;
  for (int i = wave; i < n; i += nw) {
    const int e = elist[i];
    const int s = slot[rows[e]];
    const float v = vals[e];
    const float* src = X + (size_t)cols[e] * 64;
    float* dst = acc + (size_t)s * 64;
    atomicAdd(&dst[lane], v * src[lane]);
    atomicAdd(&dst[lane + 32], v * src[lane + 32]);
  }
}

// ---------------------------------------------------------------------------
// List SpMM, 128 cols (ui -> UIN = [fused_item | item_prop] gather).
// ---------------------------------------------------------------------------
__global__ void k_spmm_list_ui(const int* __restrict__ elist, const int* __restrict__ cnt,
                               const int* __restrict__ rows, const int* __restrict__ cols,
                               const float* __restrict__ vals, const int* __restrict__ slot,
                               const float* __restrict__ FI, const float* __restrict__ IPROP,
                               float* __restrict__ acc) {
  const int lane = threadIdx.x & 31;
  const int wave = (blockIdx.x * blockDim.x + threadIdx.x) >> 5;
  const int nw = (gridDim.x * blockDim.x) >> 5;
  const int n = *cnt;
  for (int i = wave; i < n; i += nw) {
    const int e = elist[i];
    const int s = slot[rows[e]];
    const float v = vals[e];
    const float* fi = FI + (size_t)cols[e] * 64;
    const float* ip = IPROP + (size_t)cols[e] * 64;
    float* dst = acc + (size_t)s * 128;
    atomicAdd(&dst[lane], v * fi[lane]);
    atomicAdd(&dst[lane + 32], v * fi[lane + 32]);
    atomicAdd(&dst[lane + 64], v * ip[lane]);
    atomicAdd(&dst[lane + 96], v * ip[lane + 32]);
  }
}

// ---------------------------------------------------------------------------
// out = (X / (deg+EPS)) @ W ; X:(nrows,64), W:(64,64), B[k][n]=W[k*64+n]
// ---------------------------------------------------------------------------
__global__ void k_norm_gemm(const float* __restrict__ X, const float* __restrict__ deg,
                            const float* __restrict__ W, float* __restrict__ out,
                            int nrows) {
  const int wave = (blockIdx.x * blockDim.x + threadIdx.x) >> 5;
  const int lane = threadIdx.x & 31;
  const int m0 = (wave >> 2) << 4;
  if (m0 >= nrows) return;
  const int n0 = (wave & 3) << 4;
  const int half = lane >> 4;
  const int l = lane & 15;
  const int am = m0 + l;
  const float invd = 1.0f / (deg[am] + EPS_F);
  const float* xrow = X + (size_t)am * 64;
  const float* wcol = W + n0 + l;
  v8f c = {};
#pragma unroll
  for (int k0 = 0; k0 < 64; k0 += 4) {
    const int ka = k0 + half * 2;
    v2f a, b;
    a.x = xrow[ka] * invd;
    a.y = xrow[ka + 1] * invd;
    b.x = wcol[(size_t)ka * 64];
    b.y = wcol[(size_t)(ka + 1) * 64];
    c = wmma4(a, b, c);
  }
  float* orow = out + (size_t)(m0 + half * 8) * 64 + n0 + l;
#pragma unroll
  for (int v = 0; v < 8; ++v) orow[(size_t)v * 64] = c[v];
}

// ---------------------------------------------------------------------------
// Batch user update (one relation):
//   user_prop = (UPR / (ugd+EPS)) @ W            (512x64 @ 64x64, WMMA)
//   UFEAT    += user_prop + ubw * (UIN[:, :64] / ud)
// ---------------------------------------------------------------------------
__global__ void k_user_update(const float* __restrict__ UPR,
                              const float* __restrict__ UIN,
                              const int* __restrict__ buser,
                              const float* __restrict__ ugd_i,
                              const float* __restrict__ ubd,
                              const float* __restrict__ mgnn,
                              const float* __restrict__ W,
                              float* __restrict__ UFEAT, int rel) {
  const int wave = (blockIdx.x * blockDim.x + threadIdx.x) >> 5;
  const int lane = threadIdx.x & 31;
  const int m0 = (wave >> 2) << 4;
  if (m0 >= B_N) return;
  const int n0 = (wave & 3) << 4;
  const int half = lane >> 4;
  const int l = lane & 15;
  const int am = m0 + l;
  const float invd = 1.0f / (ugd_i[buser[am]] + EPS_F);
  const float* xrow = UPR + (size_t)am * 64;
  const float* wcol = W + n0 + l;
  v8f c = {};
#pragma unroll
  for (int k0 = 0; k0 < 64; k0 += 4) {
    const int ka = k0 + half * 2;
    v2f a, b;
    a.x = xrow[ka] * invd;
    a.y = xrow[ka + 1] * invd;
    b.x = wcol[(size_t)ka * 64];
    b.y = wcol[(size_t)(ka + 1) * 64];
    c = wmma4(a, b, c);
  }
  const float m0w = mgnn[0], m1w = mgnn[1], m2w = mgnn[2];
  const int col = n0 + l;
#pragma unroll
  for (int v = 0; v < 8; ++v) {
    const int row = m0 + half * 8 + v;
    const int uid = buser[row];
    const float d0 = ubd[(size_t)uid * R_N + 0];
    const float d1 = ubd[(size_t)uid * R_N + 1];
    const float d2 = ubd[(size_t)uid * R_N + 2];
    const float tw = d0 * m0w + d1 * m1w + d2 * m2w;
    const float di = (rel == 0) ? d0 : (rel == 1) ? d1 : d2;
    const float mw = (rel == 0) ? m0w : (rel == 1) ? m1w : m2w;
    const float ubw = di * mw / (tw + EPS_F);
    const float ud = di + EPS_F;
    UFEAT[(size_t)row * 64 + col] +=
        c[v] + ubw * UIN[(size_t)row * 128 + col] / ud;
  }
}

// ---------------------------------------------------------------------------
// proj = (UIN / ud) @ item_behaviour_W[rel] ; 512x128 @ 128x128
// ---------------------------------------------------------------------------
__global__ void k_proj(const float* __restrict__ UIN, const int* __restrict__ buser,
                       const float* __restrict__ ubd, const float* __restrict__ W,
                       float* __restrict__ PROJ, int rel) {
  const int wave = (blockIdx.x * blockDim.x + threadIdx.x) >> 5;
  const int lane = threadIdx.x & 31;
  const int m0 = (wave >> 3) << 4;  // 8 column tiles of 16
  if (m0 >= B_N) return;
  const int n0 = (wave & 7) << 4;
  const int half = lane >> 4;
  const int l = lane & 15;
  const int am = m0 + l;
  const float invd = 1.0f / (ubd[(size_t)buser[am] * R_N + rel] + EPS_F);
  const float* xrow = UIN + (size_t)am * 128;
  const float* wcol = W + n0 + l;
  v8f c = {};
#pragma unroll
  for (int k0 = 0; k0 < 128; k0 += 4) {
    const int ka = k0 + half * 2;
    v2f a, b;
    a.x = xrow[ka] * invd;
    a.y = xrow[ka + 1] * invd;
    b.x = wcol[(size_t)ka * 128];
    b.y = wcol[(size_t)(ka + 1) * 128];
    c = wmma4(a, b, c);
  }
  float* orow = PROJ + (size_t)(m0 + half * 8) * 128 + n0 + l;
#pragma unroll
  for (int v = 0; v < 8; ++v) orow[(size_t)v * 128] = c[v];
}

// ---------------------------------------------------------------------------
// score2[b,k] += dot(proj[slot[user[b]]], [fused_item | item_prop][item[b,k]])
// ---------------------------------------------------------------------------
__global__ void k_score2(const int* __restrict__ user, const int* __restrict__ item,
                         const int* __restrict__ slot, const float* __restrict__ PROJ,
                         const float* __restrict__ FI, const float* __restrict__ IPROP,
                         float* __restrict__ S2) {
  const int wave = (blockIdx.x * blockDim.x + threadIdx.x) >> 5;
  const int lane = threadIdx.x & 31;
  if (wave >= B_N * K_N) return;
  const int b = wave / K_N;
  const int s = slot[user[b]];
  const int it = item[wave];
  const float* p = PROJ + (size_t)s * 128;
  const float* fi = FI + (size_t)it * 64;
  const float* ip = IPROP + (size_t)it * 64;
  float sum = p[lane] * fi[lane] + p[lane + 32] * fi[lane + 32] +
              p[lane + 64] * ip[lane] + p[lane + 96] * ip[lane + 32];
  sum = wave_sum(sum);
  if (lane == 0) S2[wave] += sum;
}

// ---------------------------------------------------------------------------
// Final: score1 + LAMB*score2/R ; L2 accumulated into out[B*K].
// ---------------------------------------------------------------------------
__global__ void k_final(const int* __restrict__ user, const int* __restrict__ item,
                        const int* __restrict__ slot, const float* __restrict__ FU,
                        const float* __restrict__ UFEAT, const float* __restrict__ FI,
                        const float* __restrict__ ITF, const float* __restrict__ S2,
                        float* __restrict__ out) {
  const int wave = (blockIdx.x * blockDim.x + threadIdx.x) >> 5;
  const int lane = threadIdx.x & 31;
  if (wave >= B_N * K_N) return;
  const int b = wave / K_N;
  const int k = wave - b * K_N;
  const int uid = user[b];
  const int s = slot[uid];
  const int it = item[wave];
  const float uf0 = FU[(size_t)uid * 64 + lane];
  const float uf1 = FU[(size_t)uid * 64 + lane + 32];
  const float uf2 = UFEAT[(size_t)s * 64 + lane];
  const float uf3 = UFEAT[(size_t)s * 64 + lane + 32];
  const float tf0 = FI[(size_t)it * 64 + lane];
  const float tf1 = FI[(size_t)it * 64 + lane + 32];
  const float tf2 = ITF[(size_t)it * 64 + lane];
  const float tf3 = ITF[(size_t)it * 64 + lane + 32];
  float s1 = uf0 * tf0 + uf1 * tf1 + uf2 * tf2 + uf3 * tf3;
  float itsq = tf0 * tf0 + tf1 * tf1 + tf2 * tf2 + tf3 * tf3;
  float ufsq = uf0 * uf0 + uf1 * uf1 + uf2 * uf2 + uf3 * uf3;
  s1 = wave_sum(s1);
  itsq = wave_sum(itsq);
  ufsq = wave_sum(ufsq);
  if (lane == 0) {
    out[wave] = s1 + LAMB_F * (S2[wave] * (1.0f / R_N));
    float l2c = L2N_F * itsq;
    if (k == 0) l2c += L2N_F * (float)K_N * ufsq;
    atomicAdd(&out[B_N * K_N], l2c);
  }
}

// ---------------------------------------------------------------------------
extern "C" void kernel_launch(void* const* d_in, const int* in_sizes, int n_in,
                              void* d_out, int out_size, void* d_ws, size_t ws_size,
                              hipStream_t stream) {
  (void)in_sizes; (void)n_in; (void)out_size; (void)ws_size;
  const int* user = (const int*)d_in[0];
  const int* item = (const int*)d_in[1];
  const float* user_emb = (const float*)d_in[2];
  const float* item_emb = (const float*)d_in[3];
  const float* cat_tab = (const float*)d_in[4];
  const float* price_tab = (const float*)d_in[5];
  const float* fiW = (const float*)d_in[6];
  const float* fib = (const float*)d_in[7];
  const float* fuW = (const float*)d_in[8];
  const float* fub = (const float*)d_in[9];
  const float* mgnn = (const float*)d_in[10];
  const float* ipW = (const float*)d_in[11];
  const float* upW = (const float*)d_in[12];
  const float* ibW = (const float*)d_in[13];
  const int* category = (const int*)d_in[14];
  const int* price = (const int*)d_in[15];
  const float* hand = (const float*)d_in[16];
  const int* ui_rows = (const int*)d_in[17];
  const int* ui_cols = (const int*)d_in[18];
  const float* ui_vals = (const float*)d_in[19];
  const int* ii_rows = (const int*)d_in[20];
  const int* ii_cols = (const int*)d_in[21];
  const float* ii_vals = (const float*)d_in[22];
  const int* uu_rows = (const int*)d_in[23];
  const int* uu_cols = (const int*)d_in[24];
  const float* uu_vals = (const float*)d_in[25];
  const int* tr_rows = (const int*)d_in[26];
  const int* tr_cols = (const int*)d_in[27];
  const float* tr_vals = (const float*)d_in[28];
  const float* ubd = (const float*)d_in[29];
  const float* igd = (const float*)d_in[30];
  const float* ugd = (const float*)d_in[31];

  // Workspace layout (floats)
  float* ws = (float*)d_ws;
  size_t off = 0;
  float* FI = ws + off;    off += (size_t)I_N * 64;   // fused_item
  float* FU = ws + off;    off += (size_t)U_N * 64;   // fused_user
  float* ITF = ws + off;   off += (size_t)I_N * 64;   // item_feature (tr spmm)
  float* IACC = ws + off;  off += (size_t)I_N * 64;   // per-rel ii spmm acc
  float* IPROP = ws + off; off += (size_t)I_N * 64;   // per-rel item_prop
  float* UFEAT = ws + off; off += (size_t)B_N * 64;   // batch user_feature
  float* UIN = ws + off;   off += (size_t)B_N * 128;  // batch user_item_neigh
  float* UPR = ws + off;   off += (size_t)B_N * 64;   // batch uu spmm acc
  float* PROJ = ws + off;  off += (size_t)B_N * 128;  // proj
  float* S2 = ws + off;    off += (size_t)B_N * K_N;  // score2 accum
  int* slot = (int*)(ws + off);  off += U_N;          // batch membership map
  int* ecnt = (int*)(ws + off);  off += 1;            // compacted-edge counter
  int* elist = (int*)(ws + off);                      // compacted edge ids

  const int fiBlocks = (((I_N + 15) / 16) * 4 + 7) / 8;  // 1563
  const int fuBlocks = (((U_N + 15) / 16) * 4 + 7) / 8;  // 3125
  const int bkBlocks = (B_N * K_N + 7) / 8;              // 6400
  const int filtBlocks = (NNZ_UI_N + 255) / 256;         // 1954
  const int spmmBlocks = 2048;                           // grid-stride
  const int listBlocks = 64;                             // ~2.5K edges expected

  hipMemsetAsync(slot, 0xFF, (size_t)U_N * sizeof(int), stream);
  hipMemsetAsync(ITF, 0, (size_t)I_N * 64 * sizeof(float), stream);
  hipMemsetAsync(UFEAT, 0, (size_t)B_N * 64 * sizeof(float), stream);
  hipMemsetAsync(S2, 0, (size_t)B_N * K_N * sizeof(float), stream);
  hipMemsetAsync(d_out, 0, (size_t)(B_N * K_N + 1) * sizeof(float), stream);

  k_slot<<<(B_N + 255) / 256, 256, 0, stream>>>(user, slot, B_N);
  k_fused_item<<<fiBlocks, 256, 0, stream>>>(item_emb, cat_tab, price_tab,
                                             category, price, fiW, fib, FI, I_N);
  k_fused_user<<<fuBlocks, 256, 0, stream>>>(user_emb, hand, fuW, fub, FU, U_N);
  // item_feature = spmm(tr_cols, tr_rows, tr_vals, fused_user, I)
  k_spmm<<<spmmBlocks, 256, 0, stream>>>(tr_cols, tr_rows, tr_vals, FU, ITF,
                                         NNZ_TR_N);

  for (int i = 0; i < R_N; ++i) {
    hipMemsetAsync(IACC, 0, (size_t)I_N * 64 * sizeof(float), stream);
    hipMemsetAsync(UIN, 0, (size_t)B_N * 128 * sizeof(float), stream);
    hipMemsetAsync(UPR, 0, (size_t)B_N * 64 * sizeof(float), stream);
    // item_prop = spmm(ii, fused_item) / (igd+EPS) @ item_propagate_W[i]
    k_spmm<<<spmmBlocks, 256, 0, stream>>>(ii_rows + (size_t)i * NNZ_II_N,
                                           ii_cols + (size_t)i * NNZ_II_N,
                                           ii_vals + (size_t)i * NNZ_II_N, FI,
                                           IACC, NNZ_II_N);
    k_norm_gemm<<<fiBlocks, 256, 0, stream>>>(IACC, igd + (size_t)i * I_N,
                                              ipW + (size_t)i * 64 * 64, IPROP,
                                              I_N);
    // batch-filtered uu spmm (user_prop numerator)
    hipMemsetAsync(ecnt, 0, sizeof(int), stream);
    k_filter<<<filtBlocks, 256, 0, stream>>>(uu_rows + (size_t)i * NNZ_UU_N,
                                             slot, ecnt, elist, NNZ_UU_N);
    k_spmm_list64<<<listBlocks, 256, 0, stream>>>(
        elist, ecnt, uu_rows + (size_t)i * NNZ_UU_N,
        uu_cols + (size_t)i * NNZ_UU_N, uu_vals + (size_t)i * NNZ_UU_N, slot,
        FU, UPR);
    // batch-filtered ui spmm over [fused_item | item_prop] (single pass)
    hipMemsetAsync(ecnt, 0, sizeof(int), stream);
    k_filter<<<filtBlocks, 256, 0, stream>>>(ui_rows + (size_t)i * NNZ_UI_N,
                                             slot, ecnt, elist, NNZ_UI_N);
    k_spmm_list_ui<<<listBlocks, 256, 0, stream>>>(
        elist, ecnt, ui_rows + (size_t)i * NNZ_UI_N,
        ui_cols + (size_t)i * NNZ_UI_N, ui_vals + (size_t)i * NNZ_UI_N, slot,
        FI, IPROP, UIN);
    // user_feature += user_prop + ubw*user_neigh   (512x64 @ 64x64, WMMA)
    k_user_update<<<16, 256, 0, stream>>>(UPR, UIN, user, ugd + (size_t)i * U_N,
                                          ubd, mgnn, upW + (size_t)i * 64 * 64,
                                          UFEAT, i);
    // proj = user_item_neigh @ item_behaviour_W[i]  (512x128 @ 128x128, WMMA)
    k_proj<<<32, 256, 0, stream>>>(UIN, user, ubd,
                                   ibW + (size_t)i * 128 * 128, PROJ, i);
    k_score2<<<bkBlocks, 256, 0, stream>>>(user, item, slot, PROJ, FI, IPROP, S2);
  }

  k_final<<<bkBlocks, 256, 0, stream>>>(user, item, slot, FU, UFEAT, FI, ITF,
                                        S2, (float*)d_out);
}